// MeanAggregator_39273180955309
// MI455X (gfx1250) — compile-verified
//
#include <hip/hip_runtime.h>
#include <hip/hip_bf16.h>

typedef __attribute__((ext_vector_type(2))) float v2f;
typedef __attribute__((ext_vector_type(8))) float v8f;

#define DFEAT 128

// ---------------- pass 0: zero counters + cursors ----------------
__global__ void zero_kernel(int* __restrict__ p, int n) {
    int i = blockIdx.x * blockDim.x + threadIdx.x;
    if (i < n) p[i] = 0;
}

// ---------------- pass 1: in-degree histogram ----------------
__global__ void count_kernel(const long long* __restrict__ ei,
                             int* __restrict__ cnt, int n_edges) {
    int e = blockIdx.x * blockDim.x + threadIdx.x;
    if (e < n_edges) {
        int dst = (int)ei[n_edges + e];
        atomicAdd(&cnt[dst], 1);
    }
}

// ---------------- pass 2: exclusive prefix sum (single block) ----------------
__global__ void scan_kernel(const int* __restrict__ cnt,
                            int* __restrict__ offs, int n) {
    __shared__ int sdata[1024];
    __shared__ int carry;
    if (threadIdx.x == 0) carry = 0;
    __syncthreads();
    for (int base = 0; base < n; base += 1024) {
        int i = base + (int)threadIdx.x;
        int v = (i < n) ? cnt[i] : 0;
        sdata[threadIdx.x] = v;
        __syncthreads();
        for (int off = 1; off < 1024; off <<= 1) {
            int t = (threadIdx.x >= (unsigned)off) ? sdata[threadIdx.x - off] : 0;
            __syncthreads();
            sdata[threadIdx.x] += t;
            __syncthreads();
        }
        int incl = sdata[threadIdx.x];
        if (i < n) offs[i] = carry + incl - v;   // exclusive
        __syncthreads();                          // all carry reads done
        if (threadIdx.x == 0) carry += sdata[1023];
        __syncthreads();
    }
    if (threadIdx.x == 0) offs[n] = carry;
}

// ---------------- pass 3: scatter edges into dst-sorted buckets ----------------
// packed word: src (bits 19:0) | local dst-in-tile (bits 23:20)
__global__ void scatter_kernel(const long long* __restrict__ ei,
                               const int* __restrict__ offs,
                               int* __restrict__ cur,
                               int* __restrict__ packed, int n_edges) {
    int e = blockIdx.x * blockDim.x + threadIdx.x;
    if (e < n_edges) {
        int src = (int)ei[e];
        int dst = (int)ei[n_edges + e];
        int pos = atomicAdd(&cur[dst], 1);
        packed[offs[dst] + pos] = (src & 0xFFFFF) | ((dst & 15) << 20);
    }
}

// ---------------- pass 4: WMMA segment-sum + degree normalize ----------------
// Block = 256 threads = 8 waves; block owns one 16-dst tile, wave w owns
// feature columns [16w,16w+16). Per 4 dst-sorted edges:
//   A[m][k] = (edge k's local dst == m)   16x4 f32 selection matrix
//   B[k][n] = x[src_k][ftile*16+n]        4x16 f32 gathered features
//   D += A*B via v_wmma_f32_16x16x4_f32 (full f32 precision)

// Fast path: quad of edge words already loaded (uniform int4, no guards).
__device__ __forceinline__
v8f chunk4_fast(const float* __restrict__ x, int4 q,
                int half, int col, int fcol, v8f acc) {
    // A 16x4: VGPR0 = {K0 | K2}, VGPR1 = {K1 | K3} across half-waves
    int mx = (half ? q.z : q.x) >> 20;
    int my = (half ? q.w : q.y) >> 20;
    // B 4x16: VGPR0 = {K0 row | K2 row}, VGPR1 = {K1 row | K3 row}
    int sx = (half ? q.z : q.x) & 0xFFFFF;
    int sy = (half ? q.w : q.y) & 0xFFFFF;

    v2f a, b;
    a.x = (mx == col) ? 1.0f : 0.0f;
    a.y = (my == col) ? 1.0f : 0.0f;
    b.x = x[sx * DFEAT + fcol];
    b.y = x[sy * DFEAT + fcol];

    return __builtin_amdgcn_wmma_f32_16x16x4_f32(
        false, a, false, b, (short)0, acc, false, false);
}

// Guarded path: indices [cb, hi) valid; out-of-range K columns zeroed.
__device__ __forceinline__
v8f chunk4_guard(const float* __restrict__ x, const int* __restrict__ packed,
                 int cb, int hi, int half, int col, int fcol, v8f acc) {
    int i1 = cb + 1, i2 = cb + 2, i3 = cb + 3;
    int p0 = packed[cb];
    int p1 = packed[i1 < hi ? i1 : cb];
    int p2 = packed[i2 < hi ? i2 : cb];
    int p3 = packed[i3 < hi ? i3 : cb];
    int m0 = p0 >> 20;
    int m1 = (i1 < hi) ? (p1 >> 20) : -1;
    int m2 = (i2 < hi) ? (p2 >> 20) : -1;
    int m3 = (i3 < hi) ? (p3 >> 20) : -1;

    int mx = half ? m2 : m0;
    int my = half ? m3 : m1;
    int sx = (half ? p2 : p0) & 0xFFFFF;
    int sy = (half ? p3 : p1) & 0xFFFFF;

    v2f a, b;
    a.x = (mx == col) ? 1.0f : 0.0f;
    a.y = (my == col) ? 1.0f : 0.0f;
    b.x = x[sx * DFEAT + fcol];
    b.y = x[sy * DFEAT + fcol];

    return __builtin_amdgcn_wmma_f32_16x16x4_f32(
        false, a, false, b, (short)0, acc, false, false);
}

__global__ void agg_wmma_kernel(const float* __restrict__ x,
                                const float* __restrict__ deg,
                                const int* __restrict__ offs,
                                const int* __restrict__ packed,
                                float* __restrict__ out, int n_nodes) {
    int tile = blockIdx.x;
    int ft   = threadIdx.x >> 5;        // feature tile 0..7
    int lane = threadIdx.x & 31;
    int base = tile * 16;
    int lim  = base + 16; if (lim > n_nodes) lim = n_nodes;

    int start = offs[base];
    int end   = offs[lim];

    int half = lane >> 4;               // 0: K={0,1}, 1: K={2,3}
    int col  = lane & 15;               // N index (and M index for A rows)
    int fcol = ft * 16 + col;

    v8f acc = {};
    if (start < end) {
        // first 4-aligned chunk boundary at or after start (clamped to end)
        int cbA = (start + 3) & ~3;
        if (cbA > end) cbA = end;
        if (start < cbA)                       // guarded head (partial quad)
            acc = chunk4_guard(x, packed, start, cbA, half, col, fcol, acc);

        int cb = cbA;
        // steady state: full quads, no guards, uniform int4 edge loads
        for (; cb + 8 <= end; cb += 8) {
            __builtin_prefetch(packed + cb + 512, 0, 1);   // global_prefetch_b8
            int4 qa = *(const int4*)(packed + cb);
            int4 qb = *(const int4*)(packed + cb + 4);
            acc = chunk4_fast(x, qa, half, col, fcol, acc);
            acc = chunk4_fast(x, qb, half, col, fcol, acc);
        }
        if (cb + 4 <= end) {                   // one remaining full quad
            int4 qa = *(const int4*)(packed + cb);
            acc = chunk4_fast(x, qa, half, col, fcol, acc);
            cb += 4;
        }
        if (cb < end)                          // guarded tail (partial quad)
            acc = chunk4_guard(x, packed, cb, end, half, col, fcol, acc);
    }

    // D layout: VGPR r = rows M=r (lanes 0-15) and M=r+8 (lanes 16-31)
#pragma unroll
    for (int r = 0; r < 8; ++r) {
        int node = base + r + half * 8;
        if (node < n_nodes) {
            float dinv = __builtin_amdgcn_rcpf(deg[node]);  // v_rcp_f32
            out[node * DFEAT + fcol] = acc[r] * dinv;
        }
    }
}

extern "C" void kernel_launch(void* const* d_in, const int* in_sizes, int n_in,
                              void* d_out, int out_size, void* d_ws, size_t ws_size,
                              hipStream_t stream) {
    const float*     x   = (const float*)d_in[0];
    const long long* ei  = (const long long*)d_in[1];
    const float*     deg = (const float*)d_in[2];
    float*           out = (float*)d_out;

    int n_nodes = in_sizes[2];
    int n_edges = in_sizes[1] / 2;

    int* ws     = (int*)d_ws;
    int* cnt    = ws;                    // [n_nodes]
    int* cur    = ws + n_nodes;          // [n_nodes]
    int* offs   = ws + 2 * n_nodes;      // [n_nodes + 1]
    int* packed = ws + 3 * n_nodes + 8;  // [n_edges], 16B-aligned offset

    int zn = 2 * n_nodes;
    zero_kernel<<<(zn + 255) / 256, 256, 0, stream>>>(ws, zn);
    count_kernel<<<(n_edges + 255) / 256, 256, 0, stream>>>(ei, cnt, n_edges);
    scan_kernel<<<1, 1024, 0, stream>>>(cnt, offs, n_nodes);
    scatter_kernel<<<(n_edges + 255) / 256, 256, 0, stream>>>(ei, offs, cur, packed, n_edges);

    int tiles = (n_nodes + 15) / 16;
    agg_wmma_kernel<<<tiles, 256, 0, stream>>>(x, deg, offs, packed, out, n_nodes);
}